// diff_render_blend_20486994002611
// MI455X (gfx1250) — compile-verified
//
#include <hip/hip_runtime.h>
#include <math.h>

// ---------------------------------------------------------------------------
// diff_render_blend for gfx1250 (MI455X).
// Roofline: ~0.13 GFLOP vs ~110 MB traffic (inputs ~43 MB, fits 192 MB L2)
// => pure scatter/latency bound; no matrix structure => WMMA inapplicable.
// CDNA5-specific paths used:
//   * GLOBAL_LOAD_ASYNC_TO_LDS_B32/B128 + s_wait_asynccnt (ASYNCcnt) to stage
//     the streaming hit records into LDS (320 KB/WGP), freeing the VMEM
//     return path for the random verts/faces gathers,
//   * global_prefetch_b8 on streaming index arrays,
//   * global_atomic_max_i32 winner resolution (deterministic scatter-set),
//   * wave32-sized blocks (256 = 8 waves / WGP).
// ---------------------------------------------------------------------------

#define MAX_HITS 4
static __device__ __constant__ float PROB_INV_TEMP = 20000.0f;  // 1 / 5e-5
#define EPS_F 1e-10f
#define FAR_INIT_F 2.0f
#define BLOCK 256

// ---- CDNA5 async global->LDS support (guarded so host pass + older
//      toolchains still compile; fallback = plain cooperative LDS staging)
#if __has_builtin(__builtin_amdgcn_global_load_async_to_lds_b32) && \
    __has_builtin(__builtin_amdgcn_s_wait_asynccnt)
#define HAVE_ASYNC_LDS 1
#else
#define HAVE_ASYNC_LDS 0
#endif
#if HAVE_ASYNC_LDS && __has_builtin(__builtin_amdgcn_global_load_async_to_lds_b128)
#define HAVE_ASYNC_B128 1
#else
#define HAVE_ASYNC_B128 0
#endif

// Builtin signatures (from hipcc diagnostics):
//   b32 : (int AS1*, int AS3*, Imm offset, Imm cpol)
//   b128: (v4i AS1*, v4i AS3*, Imm offset, Imm cpol), v4i = 16B int vector
typedef int v4i __attribute__((vector_size(16)));
typedef __attribute__((address_space(1))) int*  g_i32p;
typedef __attribute__((address_space(3))) int*  l_i32p;
typedef __attribute__((address_space(1))) v4i*  g_v4ip;
typedef __attribute__((address_space(3))) v4i*  l_v4ip;

__device__ __forceinline__ g_i32p as_g32(const void* p) {
    return (g_i32p)(unsigned long long)p;   // inttoptr: global == generic repr
}
__device__ __forceinline__ l_i32p as_l32(void* p) {
    return (l_i32p)p;                       // addrspacecast generic -> LDS
}
__device__ __forceinline__ g_v4ip as_g128(const void* p) {
    return (g_v4ip)(unsigned long long)p;
}
__device__ __forceinline__ l_v4ip as_l128(void* p) {
    return (l_v4ip)p;
}

// Cooperative stage of `nwords` 32-bit words from global `g` to LDS `l`.
// Caller guarantees g and l are 16B-aligned (b128 bulk never over-reads:
// 16*(nwords>>2) <= 4*nwords).
__device__ __forceinline__ void stage_words(const void* g, void* l, int nwords) {
#if HAVE_ASYNC_LDS
    const char* gc = (const char*)g;
    char*       lc = (char*)l;
#if HAVE_ASYNC_B128
    const int nq = nwords >> 2;
    for (int t = threadIdx.x; t < nq; t += BLOCK)
        __builtin_amdgcn_global_load_async_to_lds_b128(as_g128(gc + 16 * t),
                                                       as_l128(lc + 16 * t), 0, 0);
    for (int t = (nq << 2) + threadIdx.x; t < nwords; t += BLOCK)
        __builtin_amdgcn_global_load_async_to_lds_b32(as_g32(gc + 4 * t),
                                                      as_l32(lc + 4 * t), 0, 0);
#else
    for (int t = threadIdx.x; t < nwords; t += BLOCK)
        __builtin_amdgcn_global_load_async_to_lds_b32(as_g32(gc + 4 * t),
                                                      as_l32(lc + 4 * t), 0, 0);
#endif
#else
    for (int t = threadIdx.x; t < nwords; t += BLOCK)
        ((int*)l)[t] = ((const int*)g)[t];
#endif
}

__device__ __forceinline__ void stage_join() {
#if HAVE_ASYNC_LDS
    __builtin_amdgcn_s_wait_asynccnt(0);   // my wave's async copies landed in LDS
#endif
    __syncthreads();                       // join all waves in the workgroup
}

// ---- small float3 helpers
struct V3 { float x, y, z; };
__device__ __forceinline__ V3 v3load(const float* p) { V3 r; r.x = p[0]; r.y = p[1]; r.z = p[2]; return r; }
__device__ __forceinline__ V3 v3sub(V3 a, V3 b) { V3 r; r.x = a.x - b.x; r.y = a.y - b.y; r.z = a.z - b.z; return r; }
__device__ __forceinline__ V3 v3add(V3 a, V3 b) { V3 r; r.x = a.x + b.x; r.y = a.y + b.y; r.z = a.z + b.z; return r; }
__device__ __forceinline__ V3 v3scale(float s, V3 a) { V3 r; r.x = s * a.x; r.y = s * a.y; r.z = s * a.z; return r; }
__device__ __forceinline__ float v3dot(V3 a, V3 b) { return a.x * b.x + a.y * b.y + a.z * b.z; }
__device__ __forceinline__ float safef(float x) { return (fabsf(x) < EPS_F) ? EPS_F : x; }

__device__ __forceinline__ void load_tri(const float* __restrict__ verts,
                                         const int* __restrict__ faces,
                                         int tri, V3& a, V3& b, V3& c) {
    const int i0 = faces[3 * tri + 0];
    const int i1 = faces[3 * tri + 1];
    const int i2 = faces[3 * tri + 2];
    a = v3load(verts + 3 * i0);
    b = v3load(verts + 3 * i1);
    c = v3load(verts + 3 * i2);
}

// k-level from CSR offsets: count of offsets[j] <= i, minus 1
__device__ __forceinline__ int hit_level(int i, const int* __restrict__ offs) {
    int k = -1;
#pragma unroll
    for (int j = 0; j < MAX_HITS + 1; ++j) k += (offs[j] <= i) ? 1 : 0;
    return k;
}

// ---------------------------------------------------------------------------
// Init: winners = -1 (10R ints), probs = 0 (8R floats), depth out = FAR_INIT
// ---------------------------------------------------------------------------
__global__ void k_init(int* __restrict__ win, int nwin,
                       float* __restrict__ prob, int nprob,
                       float* __restrict__ dout, int ndep) {
    const int total = nwin + nprob + ndep;
    for (int i = blockIdx.x * blockDim.x + threadIdx.x; i < total; i += gridDim.x * blockDim.x) {
        if (i < nwin)              win[i] = -1;
        else if (i < nwin + nprob) prob[i - nwin] = 0.0f;
        else                       dout[i - nwin - nprob] = FAR_INIT_F;
    }
}

// ---------------------------------------------------------------------------
// Depth phase 1: deterministic last-hit-index-wins winner per ray
// ---------------------------------------------------------------------------
__global__ void k_depth_winner(const int* __restrict__ idx_ray, int n,
                               int* __restrict__ winner) {
    for (int i = blockIdx.x * blockDim.x + threadIdx.x; i < n; i += gridDim.x * blockDim.x) {
        __builtin_prefetch(idx_ray + i + gridDim.x * blockDim.x, 0, 1);
        atomicMax(&winner[idx_ray[i]], i);
    }
}

// ---------------------------------------------------------------------------
// Depth phase 2 (tile-per-block, async-LDS staged streams):
// winning hit computes barycentric reprojection + distance to origin.
// ortho_mode==0: origin = sgrid/||sgrid||*2 ; ==1: origin = (x, +/-2, z)
// ---------------------------------------------------------------------------
__global__ void __launch_bounds__(BLOCK)
k_depth_write(const float* __restrict__ loc,
              const int* __restrict__ idx_tri,
              const int* __restrict__ idx_ray, int n,
              const float* __restrict__ verts,
              const int* __restrict__ faces,
              const float* __restrict__ sgrid,
              int ortho_mode,
              const int* __restrict__ winner,
              float* __restrict__ dout) {
    __shared__ __align__(16) float s_loc[3 * BLOCK];
    __shared__ __align__(16) int   s_tri[BLOCK];
    __shared__ __align__(16) int   s_ray[BLOCK];

    const int base = blockIdx.x * BLOCK;
    const int cnt  = (n - base < BLOCK) ? (n - base) : BLOCK;
    if (cnt <= 0) return;                           // uniform per block

    stage_words(loc + 3 * (size_t)base, s_loc, 3 * cnt);
    stage_words(idx_tri + base, s_tri, cnt);
    stage_words(idx_ray + base, s_ray, cnt);
    stage_join();

    const int t = threadIdx.x;
    const int i = base + t;
    if (t >= cnt) return;

    const int ray = s_ray[t];
    if (winner[ray] != i) return;                   // only deterministic winner writes

    V3 a, b, c;
    load_tri(verts, faces, s_tri[t], a, b, c);
    V3 p; p.x = s_loc[3 * t]; p.y = s_loc[3 * t + 1]; p.z = s_loc[3 * t + 2];

    // barycentric coords of p wrt (a,b,c)
    const V3 v0 = v3sub(b, a), v1 = v3sub(c, a), v2 = v3sub(p, a);
    const float d00 = v3dot(v0, v0), d01 = v3dot(v0, v1), d11 = v3dot(v1, v1);
    const float d20 = v3dot(v2, v0), d21 = v3dot(v2, v1);
    const float denom = safef(d00 * d11 - d01 * d01);
    const float v = (d11 * d20 - d01 * d21) / denom;
    const float w = (d00 * d21 - d01 * d20) / denom;
    const float u = 1.0f - v - w;
    const V3 nl = v3add(v3scale(u, a), v3add(v3scale(v, b), v3scale(w, c)));

    const V3 g = v3load(sgrid + 3 * ray);
    V3 o;
    if (ortho_mode) { o.x = g.x; o.y = (g.y > 0.0f) ? 2.0f : -2.0f; o.z = g.z; }
    else            { o = v3scale(2.0f * rsqrtf(v3dot(g, g)), g); }

    const V3 d = v3sub(o, nl);
    dout[ray] = sqrtf(v3dot(d, d));
}

// ---------------------------------------------------------------------------
// Silhouette phase 1: winner per (k-level, ray) slot
// ---------------------------------------------------------------------------
__global__ void k_sil_winner(const int* __restrict__ idx_ray,
                             const int* __restrict__ offsets, int n, int R,
                             int* __restrict__ winner) {
    for (int i = blockIdx.x * blockDim.x + threadIdx.x; i < n; i += gridDim.x * blockDim.x) {
        __builtin_prefetch(idx_ray + i + gridDim.x * blockDim.x, 0, 1);
        const int k = hit_level(i, offsets);
        if (k < 0 || k >= MAX_HITS) continue;
        atomicMax(&winner[k * R + idx_ray[i]], i);
    }
}

// ---------------------------------------------------------------------------
// Silhouette phase 2 (tile-per-block, async-LDS staged streams):
// winner computes Ericson point-triangle dist^2; prob = exp(-d2/temp).
// where-chain order matches the reference (later conditions override).
// ---------------------------------------------------------------------------
__global__ void __launch_bounds__(BLOCK)
k_sil_write(const float* __restrict__ loc,
            const int* __restrict__ idx_tri,
            const int* __restrict__ idx_ray,
            const int* __restrict__ offsets, int n, int R,
            const float* __restrict__ verts,
            const int* __restrict__ faces,
            const int* __restrict__ winner,
            float* __restrict__ prob) {
    __shared__ __align__(16) float s_loc[3 * BLOCK];
    __shared__ __align__(16) int   s_tri[BLOCK];
    __shared__ __align__(16) int   s_ray[BLOCK];

    const int base = blockIdx.x * BLOCK;
    const int cnt  = (n - base < BLOCK) ? (n - base) : BLOCK;
    if (cnt <= 0) return;

    stage_words(loc + 3 * (size_t)base, s_loc, 3 * cnt);
    stage_words(idx_tri + base, s_tri, cnt);
    stage_words(idx_ray + base, s_ray, cnt);
    stage_join();

    const int t = threadIdx.x;
    const int i = base + t;
    if (t >= cnt) return;

    const int k = hit_level(i, offsets);
    if (k < 0 || k >= MAX_HITS) return;
    const int flat = k * R + s_ray[t];
    if (winner[flat] != i) return;

    V3 a, b, c;
    load_tri(verts, faces, s_tri[t], a, b, c);
    V3 p; p.x = s_loc[3 * t]; p.y = s_loc[3 * t + 1]; p.z = s_loc[3 * t + 2];

    const V3 ab = v3sub(b, a), ac = v3sub(c, a), ap = v3sub(p, a);
    const float e1 = v3dot(ab, ap), e2 = v3dot(ac, ap);
    const V3 bp = v3sub(p, b);
    const float e3 = v3dot(ab, bp), e4 = v3dot(ac, bp);
    const V3 cp = v3sub(p, c);
    const float e5 = v3dot(ab, cp), e6 = v3dot(ac, cp);
    const float va = e3 * e6 - e5 * e4;
    const float vb = e5 * e2 - e1 * e6;
    const float vc = e1 * e4 - e3 * e2;

    const float denom = safef(va + vb + vc);
    const float v_in = vb / denom, w_in = vc / denom;
    V3 cl = v3add(a, v3add(v3scale(v_in, ab), v3scale(w_in, ac)));  // interior

    const float w_bc = (e4 - e3) / safef((e4 - e3) + (e5 - e6));
    if ((va <= 0.0f) && ((e4 - e3) >= 0.0f) && ((e5 - e6) >= 0.0f))
        cl = v3add(b, v3scale(w_bc, v3sub(c, b)));                  // edge BC
    if ((vb <= 0.0f) && (e2 >= 0.0f) && (e6 <= 0.0f))
        cl = v3add(a, v3scale(e2 / safef(e2 - e6), ac));            // edge AC
    if ((vc <= 0.0f) && (e1 >= 0.0f) && (e3 <= 0.0f))
        cl = v3add(a, v3scale(e1 / safef(e1 - e3), ab));            // edge AB
    if ((e6 >= 0.0f) && (e5 <= e6)) cl = c;                         // vertex C
    if ((e3 >= 0.0f) && (e4 <= e3)) cl = b;                         // vertex B
    if ((e1 <= 0.0f) && (e2 <= 0.0f)) cl = a;                       // vertex A

    const V3 df = v3sub(p, cl);
    const float d2 = v3dot(df, df);
    prob[flat] = expf(-d2 * PROB_INV_TEMP);
}

// ---------------------------------------------------------------------------
// Silhouette reduce: sil = 1 - prod_k(1 - p_k)
// ---------------------------------------------------------------------------
__global__ void k_sil_reduce(const float* __restrict__ prob, int R,
                             float* __restrict__ sout) {
    for (int ray = blockIdx.x * blockDim.x + threadIdx.x; ray < R; ray += gridDim.x * blockDim.x) {
        float q = 1.0f;
#pragma unroll
        for (int k = 0; k < MAX_HITS; ++k) q *= (1.0f - prob[k * R + ray]);
        sout[ray] = 1.0f - q;
    }
}

// ---------------------------------------------------------------------------
// Silhouette override: rays with a depth hit -> 1.0 (constant store, race-free)
// ---------------------------------------------------------------------------
__global__ void k_sil_override(const int* __restrict__ depth_idx_ray, int n,
                               float* __restrict__ sout) {
    for (int i = blockIdx.x * blockDim.x + threadIdx.x; i < n; i += gridDim.x * blockDim.x)
        sout[depth_idx_ray[i]] = 1.0f;
}

// ---------------------------------------------------------------------------
// Host launcher
// ---------------------------------------------------------------------------
extern "C" void kernel_launch(void* const* d_in, const int* in_sizes, int n_in,
                              void* d_out, int out_size, void* d_ws, size_t ws_size,
                              hipStream_t stream) {
    (void)n_in; (void)ws_size; (void)out_size;
    const float* sgrid   = (const float*)d_in[0];
    const float* verts   = (const float*)d_in[1];
    const int*   faces   = (const int*)  d_in[2];

    const float* rd_loc  = (const float*)d_in[3];
    const int*   rd_tri  = (const int*)  d_in[4];
    const int*   rd_ray  = (const int*)  d_in[5];
    const float* od_loc  = (const float*)d_in[6];
    const int*   od_tri  = (const int*)  d_in[7];
    const int*   od_ray  = (const int*)  d_in[8];

    const float* rp_loc  = (const float*)d_in[9];
    const int*   rp_tri  = (const int*)  d_in[10];
    const int*   rp_ray  = (const int*)  d_in[11];
    const int*   r_offs  = (const int*)  d_in[12];
    const float* op_loc  = (const float*)d_in[13];
    const int*   op_tri  = (const int*)  d_in[14];
    const int*   op_ray  = (const int*)  d_in[15];
    const int*   o_offs  = (const int*)  d_in[16];

    const int R  = in_sizes[0] / 3;    // 262144 rays
    const int ND = in_sizes[4];        // depth hits per pass
    const int NP = in_sizes[10];       // prob hits per pass

    // Output layout: [d_rad | d_ort | s_rad | s_ort], each R floats
    float* out   = (float*)d_out;
    float* d_rad = out + 0 * (size_t)R;
    float* d_ort = out + 1 * (size_t)R;
    float* s_rad = out + 2 * (size_t)R;
    float* s_ort = out + 3 * (size_t)R;

    // Workspace: 10R ints winners + 8R floats probs = 18R words (~18.9 MB)
    int*   winDR = (int*)d_ws;                      // R
    int*   winDO = winDR + R;                       // R
    int*   winSR = winDO + R;                       // 4R
    int*   winSO = winSR + MAX_HITS * R;            // 4R
    float* prbSR = (float*)(winSO + MAX_HITS * R);  // 4R
    float* prbSO = prbSR + MAX_HITS * R;            // 4R

    const int nwin  = (2 + 2 * MAX_HITS) * R;       // 10R
    const int nprob = 2 * MAX_HITS * R;             // 8R

    const int gInit = (nwin + nprob + 2 * R + BLOCK - 1) / BLOCK;
    const int gND   = (ND + BLOCK - 1) / BLOCK;
    const int gNP   = (NP + BLOCK - 1) / BLOCK;
    const int gR    = (R + BLOCK - 1) / BLOCK;

    // 0) re-initialize all state each call (deterministic across graph replays)
    k_init<<<gInit, BLOCK, 0, stream>>>(winDR, nwin, prbSR, nprob, d_rad, 2 * R);

    // 1) depth winners
    k_depth_winner<<<gND, BLOCK, 0, stream>>>(rd_ray, ND, winDR);
    k_depth_winner<<<gND, BLOCK, 0, stream>>>(od_ray, ND, winDO);

    // 2) depth values
    k_depth_write<<<gND, BLOCK, 0, stream>>>(rd_loc, rd_tri, rd_ray, ND, verts, faces,
                                             sgrid, /*ortho=*/0, winDR, d_rad);
    k_depth_write<<<gND, BLOCK, 0, stream>>>(od_loc, od_tri, od_ray, ND, verts, faces,
                                             sgrid, /*ortho=*/1, winDO, d_ort);

    // 3) silhouette winners
    k_sil_winner<<<gNP, BLOCK, 0, stream>>>(rp_ray, r_offs, NP, R, winSR);
    k_sil_winner<<<gNP, BLOCK, 0, stream>>>(op_ray, o_offs, NP, R, winSO);

    // 4) silhouette probabilities
    k_sil_write<<<gNP, BLOCK, 0, stream>>>(rp_loc, rp_tri, rp_ray, r_offs, NP, R,
                                           verts, faces, winSR, prbSR);
    k_sil_write<<<gNP, BLOCK, 0, stream>>>(op_loc, op_tri, op_ray, o_offs, NP, R,
                                           verts, faces, winSO, prbSO);

    // 5) combine across k-levels
    k_sil_reduce<<<gR, BLOCK, 0, stream>>>(prbSR, R, s_rad);
    k_sil_reduce<<<gR, BLOCK, 0, stream>>>(prbSO, R, s_ort);

    // 6) override silhouette where a depth hit exists
    k_sil_override<<<gND, BLOCK, 0, stream>>>(rd_ray, ND, s_rad);
    k_sil_override<<<gND, BLOCK, 0, stream>>>(od_ray, ND, s_ort);
}